// MoERouter_65403761984104
// MI455X (gfx1250) — compile-verified
//
#include <hip/hip_runtime.h>
#include <hip/hip_bf16.h>

typedef __attribute__((ext_vector_type(2))) float v2f;
typedef __attribute__((ext_vector_type(8))) float v8f;

#define NUM_EXPERTS 64

// Fragment set for one K=16 chunk (4 WMMA K-tiles of K=4 each):
// a[t]    : A-frag for k-tile t  (16x4 fp32)
// b[t][j] : B-frag for k-tile t, expert tile j (4x16 fp32)
struct Frags {
  v2f a[4];
  v2f b[4][4];
};

__device__ __forceinline__ void load_chunk(Frags& f,
                                           const float* __restrict__ tokens,
                                           const float* __restrict__ g0,
                                           const float* __restrict__ g1,
                                           const float* __restrict__ g2,
                                           const float* __restrict__ g3,
                                           uint32_t aoff, uint32_t boff,
                                           uint32_t k0) {
#pragma unroll
  for (int t = 0; t < 4; ++t) {
    const uint32_t k = k0 + 4u * (uint32_t)t;
    f.a[t]    = *(const v2f*)(tokens + (aoff + k));
    f.b[t][0] = *(const v2f*)(g0 + (boff + k));
    f.b[t][1] = *(const v2f*)(g1 + (boff + k));
    f.b[t][2] = *(const v2f*)(g2 + (boff + k));
    f.b[t][3] = *(const v2f*)(g3 + (boff + k));
  }
}

__device__ __forceinline__ void compute_chunk(const Frags& f, v8f& c0, v8f& c1,
                                              v8f& c2, v8f& c3) {
#pragma unroll
  for (int t = 0; t < 4; ++t) {
    c0 = __builtin_amdgcn_wmma_f32_16x16x4_f32(false, f.a[t], false, f.b[t][0],
                                               (short)0, c0, false, false);
    c1 = __builtin_amdgcn_wmma_f32_16x16x4_f32(false, f.a[t], false, f.b[t][1],
                                               (short)0, c1, false, false);
    c2 = __builtin_amdgcn_wmma_f32_16x16x4_f32(false, f.a[t], false, f.b[t][2],
                                               (short)0, c2, false, false);
    c3 = __builtin_amdgcn_wmma_f32_16x16x4_f32(false, f.a[t], false, f.b[t][3],
                                               (short)0, c3, false, false);
  }
}

// ---------------------------------------------------------------------------
// Kernel 1: router logits GEMM, fp32 WMMA 16x16x4, double-buffered K pipeline.
// One wave computes a 16-token x 64-expert tile (4 x v8f accumulators).
// A-frag layout: lane L -> row M = L%16, K base = 2*(L/16)  (contiguous f2).
// B-frag layout: lane L -> expert N = L%16 (gate row), same K (contiguous f2).
// ---------------------------------------------------------------------------
__global__ __launch_bounds__(256) void router_gemm_kernel(
    const float* __restrict__ tokens,  // [T, H] row-major
    const float* __restrict__ gate,    // [64, H] row-major
    float* __restrict__ logits,        // [T, 64]
    int T, int H) {
  const int lane = threadIdx.x & 31;
  const int wave = threadIdx.x >> 5;
  const int t0 = (blockIdx.x * 8 + wave) * 16;  // 8 waves/block, 16 tokens/wave
  if (t0 >= T) return;  // wave-uniform exit; EXEC stays all-1s for WMMA

  const int mrow = lane & 15;
  const int kofs = (lane >> 4) * 2;  // lanes 0-15: K={0,1}; 16-31: K={2,3}

  // 32-bit element offsets from uniform (SGPR) bases -> SADDR-form loads.
  const uint32_t aoff = (uint32_t)(t0 + mrow) * (uint32_t)H + (uint32_t)kofs;
  const uint32_t boff = (uint32_t)mrow * (uint32_t)H + (uint32_t)kofs;
  const float* g0 = gate;
  const float* g1 = gate + (size_t)16 * H;
  const float* g2 = gate + (size_t)32 * H;
  const float* g3 = gate + (size_t)48 * H;

  v8f c0 = {}, c1 = {}, c2 = {}, c3 = {};

  Frags s0, s1;
  load_chunk(s0, tokens, g0, g1, g2, g3, aoff, boff, 0);

  uint32_t k0 = 0;
  // Main pipeline: while computing one chunk, the other chunk's 20 loads are
  // in flight (waits become s_wait_loadcnt <= 20, not 0).
  for (; k0 + 32 < (uint32_t)H; k0 += 32) {
    __builtin_prefetch(tokens + (aoff + k0 + 256), 0, 0);  // speculative A-stream prefetch
    load_chunk(s1, tokens, g0, g1, g2, g3, aoff, boff, k0 + 16);
    compute_chunk(s0, c0, c1, c2, c3);
    load_chunk(s0, tokens, g0, g1, g2, g3, aoff, boff, k0 + 32);
    compute_chunk(s1, c0, c1, c2, c3);
  }
  // Epilogue: last 32 K-values (no out-of-bounds prefetch loads).
  load_chunk(s1, tokens, g0, g1, g2, g3, aoff, boff, k0 + 16);
  compute_chunk(s0, c0, c1, c2, c3);
  compute_chunk(s1, c0, c1, c2, c3);

  // C/D layout: VGPR r, lanes 0-15 -> M=r; lanes 16-31 -> M=8+r; N = lane%16.
  const int trow = t0 + 8 * (lane >> 4);
  const int ncol = lane & 15;
#pragma unroll
  for (int r = 0; r < 8; ++r) {
    float* row = logits + (size_t)(trow + r) * NUM_EXPERTS + ncol;
    row[0]  = c0[r];
    row[16] = c1[r];
    row[32] = c2[r];
    row[48] = c3[r];
  }
}

// ---------------------------------------------------------------------------
// Kernel 2: per-token top-2 + softmax-over-2 + scatter.
// One wave (32 lanes) per token; lane L owns experts {2L, 2L+1}.
// Butterfly top-2 reduction over the wave; ties resolved to the lower index
// (matches jax.lax.top_k). Reads logits in-place from the probs region and
// overwrites with the sparse probabilities; routing map written as 0.0/1.0.
// ---------------------------------------------------------------------------
__global__ __launch_bounds__(256) void router_topk_kernel(
    float* __restrict__ probs,  // [T, 64] in: logits, out: sparse probs
    float* __restrict__ rmap,   // [T, 64] routing map (0.0 / 1.0)
    int T) {
  const int lane = threadIdx.x & 31;
  const int token = blockIdx.x * 8 + (threadIdx.x >> 5);
  if (token >= T) return;

  float* row = probs + (size_t)token * NUM_EXPERTS;
  const int e0 = 2 * lane;
  const int e1 = 2 * lane + 1;
  float2 v = *(const float2*)(row + e0);

  float v1, v2;
  int i1, i2;
  if (v.x >= v.y) { v1 = v.x; i1 = e0; v2 = v.y; i2 = e1; }
  else            { v1 = v.y; i1 = e1; v2 = v.x; i2 = e0; }

#pragma unroll
  for (int off = 16; off >= 1; off >>= 1) {
    float ov1 = __shfl_xor(v1, off, 32);
    float ov2 = __shfl_xor(v2, off, 32);
    int   oi1 = __shfl_xor(i1, off, 32);
    int   oi2 = __shfl_xor(i2, off, 32);
    bool bwins = (ov1 > v1) || (ov1 == v1 && oi1 < i1);
    if (bwins) {
      bool keep = (v1 > ov2) || (v1 == ov2 && i1 < oi2);
      v2 = keep ? v1 : ov2;
      i2 = keep ? i1 : oi2;
      v1 = ov1;
      i1 = oi1;
    } else {
      bool keep = (v2 > ov1) || (v2 == ov1 && i2 < oi1);
      v2 = keep ? v2 : ov1;
      i2 = keep ? i2 : oi1;
    }
  }

  // softmax over the selected pair (v1 >= v2)
  float t = __expf(v2 - v1);
  float denom = 1.0f + t;
  float p1 = 1.0f / denom;
  float p2 = t / denom;

  float2 pout, mout;
  pout.x = (e0 == i1) ? p1 : (e0 == i2) ? p2 : 0.0f;
  pout.y = (e1 == i1) ? p1 : (e1 == i2) ? p2 : 0.0f;
  mout.x = (e0 == i1 || e0 == i2) ? 1.0f : 0.0f;
  mout.y = (e1 == i1 || e1 == i2) ? 1.0f : 0.0f;

  *(float2*)(row + e0) = pout;
  *(float2*)(rmap + (size_t)token * NUM_EXPERTS + e0) = mout;
}

extern "C" void kernel_launch(void* const* d_in, const int* in_sizes, int n_in,
                              void* d_out, int out_size, void* d_ws, size_t ws_size,
                              hipStream_t stream) {
  const float* tokens = (const float*)d_in[0];  // [S, B, H] fp32 -> flat [T, H]
  const float* gate   = (const float*)d_in[1];  // [64, H] fp32
  (void)n_in; (void)d_ws; (void)ws_size; (void)out_size;

  const int E = NUM_EXPERTS;
  const int H = in_sizes[1] / E;                    // 4096
  const int T = (int)((long long)in_sizes[0] / H);  // 16384

  float* probs = (float*)d_out;          // [T, 64]
  float* rmap  = probs + (size_t)T * E;  // [T, 64]

  dim3 block(256);
  dim3 grid1((unsigned)((T + 127) / 128));  // 8 waves/block * 16 tokens/wave
  router_gemm_kernel<<<grid1, block, 0, stream>>>(tokens, gate, probs, T, H);

  dim3 grid2((unsigned)((T + 7) / 8));  // one wave per token
  router_topk_kernel<<<grid2, block, 0, stream>>>(probs, rmap, T);
}